// GATEncoder_38491496906889
// MI455X (gfx1250) — compile-verified
//
#include <hip/hip_runtime.h>
#include <stdint.h>

typedef float v2f __attribute__((ext_vector_type(2)));
typedef float v8f __attribute__((ext_vector_type(8)));

#define N_NODES 50000
#define N_EDGES 800000
#define ETOT    (N_EDGES + N_NODES)   // reference appends one self-loop per node
#define IN_CH   64
#define F1      256                   // heads*channels layer 1
#define H1      4
#define C1      64
#define F2      32                    // layer 2: 1 head, 32 channels
#define NEG_SLOPE 0.2f

// ---------------------------------------------------------------- utilities
__global__ void zero_f32(float* __restrict__ p, int n) {
  int i = blockIdx.x * blockDim.x + threadIdx.x;
  if (i < n) p[i] = 0.0f;
}
__global__ void zero_u32(unsigned* __restrict__ p, int n) {
  int i = blockIdx.x * blockDim.x + threadIdx.x;
  if (i < n) p[i] = 0u;
}

__device__ __forceinline__ void edge_sd(const int* __restrict__ ei, int e, int& s, int& d) {
  if (e < N_EDGES) { s = ei[e]; d = ei[N_EDGES + e]; }
  else             { s = d = e - N_EDGES; }          // self loop
}

// monotone float<->uint key for atomicMax-based segment max
__device__ __forceinline__ unsigned f2key(float f) {
  unsigned b = __float_as_uint(f);
  return (b & 0x80000000u) ? ~b : (b | 0x80000000u);
}
__device__ __forceinline__ float key2f(unsigned k) {
  if (k == 0u) return 0.0f;                          // untouched -> finite-guard 0
  unsigned b = (k & 0x80000000u) ? (k & 0x7fffffffu) : ~k;
  return __uint_as_float(b);
}

__device__ __forceinline__ float leaky(float x) {
  return x > 0.0f ? x : NEG_SLOPE * x;
}

// ---------------------------------------------------------------- fp32 WMMA GEMM
// C[M,N] = A[M,K] * B[K,N], row-major fp32; M%16==0, N%(16*NT)==0, K%4==0.
// One wave32 computes a 16 x (16*NT) strip: the A fragment is loaded once per
// k-step and fed to NT independent V_WMMA_F32_16X16X4_F32 accumulator chains.
template <int NT>
__global__ void wmma_gemm_f32(const float* __restrict__ A, const float* __restrict__ B,
                              float* __restrict__ C, int M, int K, int N, int tiles) {
  const int lane = threadIdx.x & 31;
  const int tile = blockIdx.x * (blockDim.x >> 5) + (threadIdx.x >> 5);
  if (tile >= tiles) return;
  const int tilesN = N / (16 * NT);
  const int tm  = tile / tilesN;
  const int tn0 = (tile % tilesN) * NT;

  const int rc   = lane & 15;         // A row / B,C column within tile
  const int hi   = lane >> 4;         // half-select
  const int koff = hi << 1;           // lanes 16-31 carry K+2,K+3

  const float* Ap = A + (size_t)(tm * 16 + rc) * K + koff;
  const float* Bp = B + (size_t)koff * N + tn0 * 16 + rc;

  v8f acc[NT];
#pragma unroll
  for (int t = 0; t < NT; ++t) acc[t] = (v8f){};

  for (int k0 = 0; k0 < K; k0 += 4) {
    v2f a;
    a.x = Ap[k0];
    a.y = Ap[k0 + 1];
#pragma unroll
    for (int t = 0; t < NT; ++t) {
      v2f b;
      b.x = Bp[(size_t)k0 * N + t * 16];
      b.y = Bp[(size_t)(k0 + 1) * N + t * 16];
      acc[t] = __builtin_amdgcn_wmma_f32_16x16x4_f32(false, a, false, b, (short)0,
                                                     acc[t], false, false);
    }
  }
  // D 16x16 f32: lane&15 = N, VGPR r = row r (lanes 0-15) / row r+8 (lanes 16-31)
  float* Cp = C + (size_t)(tm * 16 + hi * 8) * N + tn0 * 16 + rc;
#pragma unroll
  for (int t = 0; t < NT; ++t)
#pragma unroll
    for (int r = 0; r < 8; ++r) Cp[(size_t)r * N + t * 16] = acc[t][r];
}

// ---------------------------------------------------------------- attention dots
// as[i,h] = sum_c h[i,h,c]*att_src[h,c] ; ad likewise. One thread per (node,head).
template <int CH>
__global__ void att_dots(const float* __restrict__ h, const float* __restrict__ att_s,
                         const float* __restrict__ att_d, float* __restrict__ as,
                         float* __restrict__ ad, int n, int heads) {
  int idx = blockIdx.x * blockDim.x + threadIdx.x;
  if (idx >= n * heads) return;
  int node = idx / heads, hd = idx % heads;
  const float* hp = h + (size_t)node * heads * CH + (size_t)hd * CH;
  const float* sp = att_s + (size_t)hd * CH;
  const float* dp = att_d + (size_t)hd * CH;
  float s = 0.0f, d = 0.0f;
#pragma unroll 4
  for (int c = 0; c < CH; ++c) { float v = hp[c]; s += v * sp[c]; d += v * dp[c]; }
  as[idx] = s;
  ad[idx] = d;
}

// ---------------------------------------------------------------- softmax pass 1
// Compute + materialize logits, atomicMax the per-(dst,head) running max.
template <int H>
__global__ void edge_logit_max(const int* __restrict__ ei, const float* __restrict__ as,
                               const float* __restrict__ ad, float* __restrict__ elog,
                               unsigned* __restrict__ lkey) {
  int idx = blockIdx.x * blockDim.x + threadIdx.x;   // e*H + h
  if (idx >= ETOT * H) return;
  int e = idx / H, h = idx % H;
  int s, d; edge_sd(ei, e, s, d);
  float lg = leaky(as[s * H + h] + ad[d * H + h]);
  elog[idx] = lg;
  atomicMax(&lkey[d * H + h], f2key(lg));
}

// ---------------------------------------------------------------- softmax pass 2
// ex = exp(logit - max); stored in place over the logit; denominator atomicAdd.
template <int H>
__global__ void edge_expsum(const int* __restrict__ ei, float* __restrict__ elog,
                            const unsigned* __restrict__ lkey, float* __restrict__ denom) {
  int idx = blockIdx.x * blockDim.x + threadIdx.x;
  if (idx >= ETOT * H) return;
  int e = idx / H, h = idx % H;
  int s, d; edge_sd(ei, e, s, d);
  float m  = key2f(lkey[d * H + h]);
  float ex = expf(elog[idx] - m);
  elog[idx] = ex;
  atomicAdd(&denom[d * H + h], ex);
}

// ---------------------------------------------------------------- pass 3: aggregate
// One wave per edge; each lane owns H*CH/32 consecutive channels (within one head).
template <int H, int CH>
__global__ void edge_aggr(const int* __restrict__ ei, const float* __restrict__ elog,
                          const float* __restrict__ denom, const float* __restrict__ h,
                          float* __restrict__ out) {
  int wave = blockIdx.x * (blockDim.x >> 5) + (threadIdx.x >> 5);
  int lane = threadIdx.x & 31;
  if (wave >= ETOT) return;
  int s, d; edge_sd(ei, wave, s, d);
  constexpr int F   = H * CH;
  constexpr int per = F / 32;             // 8 channels/lane (L1), 1 (L2)
  const int c0 = lane * per;
  const int hd = c0 / CH;
  float alpha = elog[wave * H + hd] / (denom[d * H + hd] + 1e-16f);
  const float* hp = h + (size_t)s * F + c0;
  float*       op = out + (size_t)d * F + c0;
  __builtin_prefetch(hp, 0, 1);
#pragma unroll
  for (int j = 0; j < per; ++j) atomicAdd(&op[j], alpha * hp[j]);
}

// ---------------------------------------------------------------- epilogues
__global__ void bias_elu(float* __restrict__ h, const float* __restrict__ bias, int n, int F) {
  int i = blockIdx.x * blockDim.x + threadIdx.x;
  if (i >= n) return;
  float v = h[i] + bias[i % F];
  h[i] = v > 0.0f ? v : (expf(v) - 1.0f);
}
__global__ void bias_add(float* __restrict__ o, const float* __restrict__ bias, int n, int F) {
  int i = blockIdx.x * blockDim.x + threadIdx.x;
  if (i >= n) return;
  o[i] += bias[i % F];
}

// ---------------------------------------------------------------- launcher
extern "C" void kernel_launch(void* const* d_in, const int* in_sizes, int n_in,
                              void* d_out, int out_size, void* d_ws, size_t ws_size,
                              hipStream_t stream) {
  const float* x   = (const float*)d_in[0];
  const int*   ei  = (const int*)d_in[1];     // edge_index [2,E]
  const float* W1  = (const float*)d_in[2];
  const float* as1 = (const float*)d_in[3];
  const float* ad1 = (const float*)d_in[4];
  const float* b1  = (const float*)d_in[5];
  const float* W2  = (const float*)d_in[6];
  const float* as2 = (const float*)d_in[7];
  const float* ad2 = (const float*)d_in[8];
  const float* b2  = (const float*)d_in[9];
  float* out = (float*)d_out;                 // [N, 32]

  // workspace carve-up (256B aligned)
  char* w = (char*)d_ws;
  auto carve = [&](size_t bytes) { char* p = w; w += (bytes + 255) & ~(size_t)255; return p; };
  float*    h1pre = (float*)   carve(sizeof(float) * N_NODES * F1);
  float*    h1    = (float*)   carve(sizeof(float) * N_NODES * F1);
  float*    h2pre = (float*)   carve(sizeof(float) * N_NODES * F2);
  float*    elog1 = (float*)   carve(sizeof(float) * (size_t)ETOT * H1);
  float*    elog2 = (float*)   carve(sizeof(float) * (size_t)ETOT);
  float*    asrc1 = (float*)   carve(sizeof(float) * N_NODES * H1);
  float*    adst1 = (float*)   carve(sizeof(float) * N_NODES * H1);
  float*    den1  = (float*)   carve(sizeof(float) * N_NODES * H1);
  unsigned* lk1   = (unsigned*)carve(sizeof(unsigned) * N_NODES * H1);
  float*    asrc2 = (float*)   carve(sizeof(float) * N_NODES);
  float*    adst2 = (float*)   carve(sizeof(float) * N_NODES);
  float*    den2  = (float*)   carve(sizeof(float) * N_NODES);
  unsigned* lk2   = (unsigned*)carve(sizeof(unsigned) * N_NODES);

  const int BLK = 256;
  auto cdiv = [](int a, int b) { return (a + b - 1) / b; };

  // --- init accumulators (d_out/ws are poisoned; atomics need zeros)
  zero_f32<<<cdiv(N_NODES * F1, BLK), BLK, 0, stream>>>(h1, N_NODES * F1);
  zero_f32<<<cdiv(N_NODES * H1, BLK), BLK, 0, stream>>>(den1, N_NODES * H1);
  zero_u32<<<cdiv(N_NODES * H1, BLK), BLK, 0, stream>>>(lk1, N_NODES * H1);
  zero_f32<<<cdiv(N_NODES * F2, BLK), BLK, 0, stream>>>(out, N_NODES * F2);
  zero_f32<<<cdiv(N_NODES, BLK), BLK, 0, stream>>>(den2, N_NODES);
  zero_u32<<<cdiv(N_NODES, BLK), BLK, 0, stream>>>(lk2, N_NODES);

  // --- layer 1: h1pre = x @ W1  (50000x64 * 64x256), 16x64 strip per wave
  {
    int tiles = (N_NODES / 16) * (F1 / 64);
    wmma_gemm_f32<4><<<cdiv(tiles, BLK / 32), BLK, 0, stream>>>(x, W1, h1pre,
                                                                N_NODES, IN_CH, F1, tiles);
  }
  att_dots<C1><<<cdiv(N_NODES * H1, BLK), BLK, 0, stream>>>(h1pre, as1, ad1,
                                                            asrc1, adst1, N_NODES, H1);
  edge_logit_max<H1><<<cdiv(ETOT * H1, BLK), BLK, 0, stream>>>(ei, asrc1, adst1, elog1, lk1);
  edge_expsum<H1><<<cdiv(ETOT * H1, BLK), BLK, 0, stream>>>(ei, elog1, lk1, den1);
  edge_aggr<H1, C1><<<cdiv(ETOT, BLK / 32), BLK, 0, stream>>>(ei, elog1, den1, h1pre, h1);
  bias_elu<<<cdiv(N_NODES * F1, BLK), BLK, 0, stream>>>(h1, b1, N_NODES * F1, F1);

  // --- layer 2: h2pre = h1 @ W2  (50000x256 * 256x32), 16x32 strip per wave
  {
    int tiles = (N_NODES / 16) * (F2 / 32);
    wmma_gemm_f32<2><<<cdiv(tiles, BLK / 32), BLK, 0, stream>>>(h1, W2, h2pre,
                                                                N_NODES, F1, F2, tiles);
  }
  att_dots<F2><<<cdiv(N_NODES, BLK), BLK, 0, stream>>>(h2pre, as2, ad2,
                                                       asrc2, adst2, N_NODES, 1);
  edge_logit_max<1><<<cdiv(ETOT, BLK), BLK, 0, stream>>>(ei, asrc2, adst2, elog2, lk2);
  edge_expsum<1><<<cdiv(ETOT, BLK), BLK, 0, stream>>>(ei, elog2, lk2, den2);
  edge_aggr<1, F2><<<cdiv(ETOT, BLK / 32), BLK, 0, stream>>>(ei, elog2, den2, h2pre, out);
  bias_add<<<cdiv(N_NODES * F2, BLK), BLK, 0, stream>>>(out, b2, N_NODES * F2, F2);
}